// Generator_65807488909717
// MI455X (gfx1250) — compile-verified
//
#include <hip/hip_runtime.h>
#include <hip/hip_bf16.h>
#include <cstdint>

typedef __attribute__((ext_vector_type(2))) float v2f;
typedef __attribute__((ext_vector_type(8))) float v8f;
typedef __attribute__((ext_vector_type(4))) unsigned int u32x4;
typedef __attribute__((ext_vector_type(4))) int i32x4;
typedef __attribute__((ext_vector_type(8))) int i32x8;

#define Bb   16
#define Ss   200
#define Hh   400
#define Vv   25000
#define Tt   8
#define NSs  30
#define NBb  3
#define BS   (NSs * Bb)     // 480
#define G3   (3 * Hh)       // 1200
#define NPAD 25008          // V padded to 16

#if __has_builtin(__builtin_amdgcn_tensor_load_to_lds)
#define HAVE_TDM 1
#else
#define HAVE_TDM 0
#endif

// ---------------- TDM: contiguous global -> LDS copy (one descriptor) --------
#if HAVE_TDM
__device__ __forceinline__ void tdm_load_contig(const float* gsrc, unsigned int lds_off,
                                                unsigned int nelem) {
    unsigned long long ga = (unsigned long long)(uintptr_t)gsrc;
    // D# group0: count=1 | lds_addr | global_addr | type=2 in bits[127:126]
    u32x4 g0;
    g0.x = 1u;
    g0.y = lds_off;
    g0.z = (unsigned int)(ga & 0xffffffffu);
    g0.w = (unsigned int)((ga >> 32) & 0x01ffffffu) | 0x80000000u;
    // D# group1: data_size=4B; tile = nelem x 1; tensor_dim0 = nelem; stride0 = nelem
    i32x8 g1;
    g1[0] = 0x20000;                                   // data_size=2 -> 4 bytes
    g1[1] = (int)((nelem & 0xffffu) << 16);            // tensor_dim0[15:0]
    g1[2] = (int)((nelem >> 16) & 0xffffu) | (1 << 16);// tensor_dim0[31:16] | tensor_dim1=1
    g1[3] = (int)((nelem & 0xffffu) << 16);            // tile_dim0
    g1[4] = 1;                                         // tile_dim1=1, tile_dim2=0
    g1[5] = (int)nelem;                                // tensor_dim0_stride low32
    g1[6] = 0;
    g1[7] = 0;
    i32x4 z4 = {0, 0, 0, 0};
#if defined(__clang_major__) && (__clang_major__ >= 23)
    i32x8 z8 = {0, 0, 0, 0, 0, 0, 0, 0};
    __builtin_amdgcn_tensor_load_to_lds(g0, g1, z4, z4, z8, 0);
#else
    __builtin_amdgcn_tensor_load_to_lds(g0, g1, z4, z4, 0);
#endif
}
#endif

// ---------------- block reductions ----------------
__device__ __forceinline__ float block_reduce_max(float v, float* rbuf) {
    int tid = threadIdx.x;
    rbuf[tid] = v; __syncthreads();
    for (int off = 128; off > 0; off >>= 1) {
        if (tid < off) rbuf[tid] = fmaxf(rbuf[tid], rbuf[tid + off]);
        __syncthreads();
    }
    float r = rbuf[0]; __syncthreads();
    return r;
}
__device__ __forceinline__ float block_reduce_sum(float v, float* rbuf) {
    int tid = threadIdx.x;
    rbuf[tid] = v; __syncthreads();
    for (int off = 128; off > 0; off >>= 1) {
        if (tid < off) rbuf[tid] += rbuf[tid + off];
        __syncthreads();
    }
    float r = rbuf[0]; __syncthreads();
    return r;
}
__device__ __forceinline__ float sigmoidf(float x) { return 1.f / (1.f + expf(-x)); }

// ---------------- WMMA f32 GEMM: OUT[m,n] = sum_k X[m,k]*W[n,k] (+bias[n]) ----
// K fixed at 400. MT m-tiles (16 rows each) staged in LDS per block (CDNA5 WGP
// LDS is 320KB, so 80 rows = 128KB is fine); one W float2 load feeds MT WMMAs.
// Grid: (M/(16*MT), ceil(Ntiles/8)), 256 threads = 8 waves, 1 N-tile per wave.
template <int NN, int LDO, int NCLAMP, bool BIAS, int MT>
__global__ void gemm_nt_wmma(const float* __restrict__ X, const float* __restrict__ W,
                             const float* __restrict__ bias, float* __restrict__ OUT) {
    __shared__ float shx[MT * 16 * Hh];                // A tiles (MT*25.6 KB)
    const int tid = threadIdx.x;
    const int m0  = blockIdx.x * (16 * MT);
    // X is densely packed (row stride == Hh) so the MT*16-row tile is one
    // contiguous block: stage it with a single Tensor Data Mover descriptor.
    // readfirstlane makes the gate scalar -> scalar branch -> exactly one wave
    // issues the TDM (TENSOR ops ignore EXEC, so predication would duplicate).
#if HAVE_TDM
    if (__builtin_amdgcn_readfirstlane((int)threadIdx.x) == 0) {
        tdm_load_contig(X + (size_t)m0 * Hh, (unsigned int)(uintptr_t)shx, MT * 16 * Hh);
        __builtin_amdgcn_s_wait_tensorcnt(0);
    }
#else
    {
        const float* xb = X + (size_t)m0 * Hh;
        for (int idx = tid; idx < MT * 16 * Hh; idx += 256) shx[idx] = xb[idx];
    }
#endif
    __syncthreads();

    const int wave = tid >> 5;
    const int lane = tid & 31;
    const int half = lane >> 4;      // 0: lanes 0-15 (K=0,1) | 1: lanes 16-31 (K=2,3)
    const int l16  = lane & 15;
    const int n0   = (blockIdx.y * 8 + wave) * 16;
    if (n0 >= NN) return;            // wave-uniform: EXEC stays all-ones inside

    const int ncol = n0 + l16;
    const int nrow = ncol < NCLAMP ? ncol : (NCLAMP - 1); // clamp partial last tile
    const float* wr = W + (size_t)nrow * Hh;              // B[k][n] = W[n][k]

    v8f acc[MT];
    #pragma unroll
    for (int mt = 0; mt < MT; ++mt) acc[mt] = (v8f){};

    #pragma unroll 4
    for (int k = 0; k < Hh; k += 4) {
        const int ko = k + half * 2;
        v2f b; b.x = wr[ko]; b.y = wr[ko + 1];
        #pragma unroll
        for (int mt = 0; mt < MT; ++mt) {
            const float* xr = &shx[(mt * 16 + l16) * Hh];
            v2f a; a.x = xr[ko]; a.y = xr[ko + 1];
            acc[mt] = __builtin_amdgcn_wmma_f32_16x16x4_f32(false, a, false, b,
                                                            (short)0, acc[mt],
                                                            false, false);
        }
    }
    // C/D layout: VGPR j -> M = j + 8*half, N = lane&15
    float bv = 0.f;
    if constexpr (BIAS) bv = bias[ncol];
    #pragma unroll
    for (int mt = 0; mt < MT; ++mt) {
        float* orow = OUT + (size_t)(m0 + mt * 16 + half * 8) * LDO + ncol;
        #pragma unroll
        for (int j = 0; j < 8; ++j) {
            orow[(size_t)j * LDO] = acc[mt][j] + bv;
        }
    }
}

// ---------------- step-0 state init: h0 tile + slot embedding input ----------
__global__ void init_kernel(const float* __restrict__ enc_hid,
                            const float* __restrict__ slot_emb_W,
                            const int* __restrict__ dom, const int* __restrict__ sl,
                            float* __restrict__ hid, float* __restrict__ x) {
    int i = blockIdx.x, tid = threadIdx.x;
    int slot = i / Bb, b = i - slot * Bb;
    int d = dom[slot], s2 = sl[slot];
    for (int h = tid; h < Hh; h += 256) {
        hid[(size_t)i * Hh + h] = enc_hid[b * Hh + h];
        x[(size_t)i * Hh + h]   = slot_emb_W[d * Hh + h] + slot_emb_W[s2 * Hh + h];
    }
}

// ---------------- teacher forcing gather: x = emb_W[target[b,slot,t-1]] ------
__global__ void xgather_kernel(const int* __restrict__ tb, const float* __restrict__ emb,
                               float* __restrict__ x, int tprev) {
    int i = blockIdx.x, tid = threadIdx.x;
    int slot = i / Bb, b = i - slot * Bb;
    int tgt = tb[(b * NSs + slot) * Tt + tprev];
    for (int h = tid; h < Hh; h += 256)
        x[(size_t)i * Hh + h] = emb[(size_t)tgt * Hh + h];
}

// ---------------- GRU gate fusion (hid updated in place) ---------------------
__global__ void gru_gate_kernel(const float* __restrict__ gx, const float* __restrict__ gh,
                                float* __restrict__ hid) {
    int idx = blockIdx.x * blockDim.x + threadIdx.x;
    if (idx >= BS * Hh) return;
    int i = idx / Hh, h = idx - i * Hh;
    const float* gxr = gx + (size_t)i * G3;
    const float* ghr = gh + (size_t)i * G3;
    float r = sigmoidf(gxr[h] + ghr[h]);
    float z = sigmoidf(gxr[Hh + h] + ghr[Hh + h]);
    float n = tanhf(gxr[2 * Hh + h] + r * ghr[2 * Hh + h]);
    hid[idx] = (1.f - z) * n + z * hid[idx];
}

// ---------------- attention: scores -> masked softmax -> ctx -----------------
__global__ void attn_kernel(const float* __restrict__ hid, const float* __restrict__ enc,
                            const int* __restrict__ story,
                            float* __restrict__ prob, float* __restrict__ ctx) {
    __shared__ float sh_h[Hh];
    __shared__ float sh_sc[Ss];
    __shared__ float rbuf[256];
    int i = blockIdx.x, tid = threadIdx.x;
    int b = i & (Bb - 1);
    for (int h = tid; h < Hh; h += 256) sh_h[h] = hid[(size_t)i * Hh + h];
    __syncthreads();
    for (int s = tid; s < Ss; s += 256) {
        const float4* e4 = reinterpret_cast<const float4*>(enc + ((size_t)b * Ss + s) * Hh);
        const float4* h4 = reinterpret_cast<const float4*>(sh_h);
        float acc = 0.f;
        #pragma unroll 4
        for (int q = 0; q < Hh / 4; ++q) {
            float4 e = e4[q], hh = h4[q];
            acc += e.x * hh.x + e.y * hh.y + e.z * hh.z + e.w * hh.w;
        }
        sh_sc[s] = (story[b * Ss + s] == 0) ? -3.0e38f : acc;
    }
    __syncthreads();
    float lm = -3.0e38f;
    for (int s = tid; s < Ss; s += 256) lm = fmaxf(lm, sh_sc[s]);
    float mx = block_reduce_max(lm, rbuf);
    float ls = 0.f;
    for (int s = tid; s < Ss; s += 256) { float e = expf(sh_sc[s] - mx); sh_sc[s] = e; ls += e; }
    float sum = block_reduce_sum(ls, rbuf);
    float inv = 1.f / sum;
    for (int s = tid; s < Ss; s += 256) {
        float p = sh_sc[s] * inv;
        sh_sc[s] = p;
        prob[(size_t)i * Ss + s] = p;
    }
    __syncthreads();
    for (int h = tid; h < Hh; h += 256) {
        float acc = 0.f;
        for (int s = 0; s < Ss; ++s) acc += sh_sc[s] * enc[((size_t)b * Ss + s) * Hh + h];
        ctx[(size_t)i * Hh + h] = acc;
    }
}

// ---------------- switch gate: sigmoid([hid|ctx|x] . w + b) ------------------
__global__ void switch_kernel(const float* __restrict__ hid, const float* __restrict__ ctx,
                              const float* __restrict__ x, const float* __restrict__ wr,
                              const float* __restrict__ br, float* __restrict__ sw) {
    __shared__ float rbuf[256];
    int i = blockIdx.x, tid = threadIdx.x;
    float local = 0.f;
    for (int h = tid; h < Hh; h += 256) {
        local += hid[(size_t)i * Hh + h] * wr[h]
               + ctx[(size_t)i * Hh + h] * wr[Hh + h]
               + x[(size_t)i * Hh + h]   * wr[2 * Hh + h];
    }
    float sum = block_reduce_sum(local, rbuf);
    if (tid == 0) sw[i] = sigmoidf(sum + br[0]);
}

// ---------------- vocab softmax * switch -> d_out ----------------------------
__global__ void vocab_softmax_kernel(const float* __restrict__ logits,
                                     const float* __restrict__ sw,
                                     float* __restrict__ out, int t) {
    __shared__ float rbuf[256];
    int i = blockIdx.x, tid = threadIdx.x;
    const float* lr = logits + (size_t)i * NPAD;
    float lm = -3.0e38f;
    for (int v = tid; v < Vv; v += 256) lm = fmaxf(lm, lr[v]);
    float mx = block_reduce_max(lm, rbuf);
    float ls = 0.f;
    for (int v = tid; v < Vv; v += 256) ls += expf(lr[v] - mx);
    float sum = block_reduce_sum(ls, rbuf);
    float scale = sw[i] / sum;
    float* orow = out + ((size_t)i * Tt + t) * Vv;
    for (int v = tid; v < Vv; v += 256) orow[v] = expf(lr[v] - mx) * scale;
}

// ---------------- pointer (copy) distribution scatter-add --------------------
__global__ void scatter_kernel(const float* __restrict__ prob, const int* __restrict__ story,
                               const float* __restrict__ sw, float* __restrict__ out, int t) {
    int idx = blockIdx.x * blockDim.x + threadIdx.x;
    if (idx >= BS * Ss) return;
    int i = idx / Ss, s = idx - i * Ss;
    int b = i & (Bb - 1);
    int v = story[b * Ss + s];
    float val = (1.f - sw[i]) * prob[idx];   // prob==0 at pad positions
    atomicAdd(out + ((size_t)i * Tt + t) * Vv + v, val);
}

// ---------------- gate head (step 0 ctx only) --------------------------------
__global__ void gate_kernel(const float* __restrict__ ctx, const float* __restrict__ wg,
                            const float* __restrict__ bg, float* __restrict__ outg) {
    __shared__ float rbuf[256];
    int i = blockIdx.x, tid = threadIdx.x;
    for (int j = 0; j < NBb; ++j) {
        float local = 0.f;
        for (int h = tid; h < Hh; h += 256) local += ctx[(size_t)i * Hh + h] * wg[j * Hh + h];
        float sum = block_reduce_sum(local, rbuf);
        if (tid == 0) outg[i * NBb + j] = sum + bg[j];
        __syncthreads();
    }
}

extern "C" void kernel_launch(void* const* d_in, const int* in_sizes, int n_in,
                              void* d_out, int out_size, void* d_ws, size_t ws_size,
                              hipStream_t stream) {
    const float* enc_hid  = (const float*)d_in[0];
    const float* enc_out  = (const float*)d_in[1];
    const int*   story    = (const int*)d_in[2];
    const int*   tb       = (const int*)d_in[3];
    const int*   dom      = (const int*)d_in[4];
    const int*   sl       = (const int*)d_in[5];
    const float* emb      = (const float*)d_in[6];
    const float* slot_emb = (const float*)d_in[7];
    const float* Wih      = (const float*)d_in[8];
    const float* Whh      = (const float*)d_in[9];
    const float* bih      = (const float*)d_in[10];
    const float* bhh      = (const float*)d_in[11];
    const float* wr       = (const float*)d_in[12];
    const float* br       = (const float*)d_in[13];
    const float* wg       = (const float*)d_in[14];
    const float* bg       = (const float*)d_in[15];
    float* out = (float*)d_out;

    // workspace carve (~55 MB of f32)
    float* p      = (float*)d_ws;
    float* hid    = p; p += (size_t)BS * Hh;
    float* x      = p; p += (size_t)BS * Hh;
    float* gx     = p; p += (size_t)BS * G3;
    float* gh     = p; p += (size_t)BS * G3;
    float* prob   = p; p += (size_t)BS * Ss;
    float* ctx    = p; p += (size_t)BS * Hh;
    float* sw     = p; p += (size_t)BS;
    float* logits = p; p += (size_t)BS * NPAD;

    constexpr int MT = 5;                                    // m-tiles per block (128KB LDS)
    const dim3 gGate(BS / (16 * MT), (G3 / 16 + 7) / 8);     // (6, 10)
    const dim3 gVoc(BS / (16 * MT), ((Vv + 15) / 16 + 7) / 8); // (6, 196)

    for (int t = 0; t < Tt; ++t) {
        if (t == 0)
            init_kernel<<<BS, 256, 0, stream>>>(enc_hid, slot_emb, dom, sl, hid, x);
        else
            xgather_kernel<<<BS, 256, 0, stream>>>(tb, emb, x, t - 1);

        // GRU: gx = x @ Wih^T + bih ; gh = hid @ Whh^T + bhh  (WMMA f32)
        gemm_nt_wmma<G3, G3, G3, true, MT><<<gGate, 256, 0, stream>>>(x,   Wih, bih, gx);
        gemm_nt_wmma<G3, G3, G3, true, MT><<<gGate, 256, 0, stream>>>(hid, Whh, bhh, gh);
        gru_gate_kernel<<<(BS * Hh + 255) / 256, 256, 0, stream>>>(gx, gh, hid);

        attn_kernel<<<BS, 256, 0, stream>>>(hid, enc_out, story, prob, ctx);
        if (t == 0)
            gate_kernel<<<BS, 256, 0, stream>>>(ctx, wg, bg,
                                                out + (size_t)NSs * Bb * Tt * Vv);
        switch_kernel<<<BS, 256, 0, stream>>>(hid, ctx, x, wr, br, sw);

        // vocab projection (WMMA f32), then softmax*switch write, then copy scatter
        gemm_nt_wmma<Vv, NPAD, Vv, false, MT><<<gVoc, 256, 0, stream>>>(hid, emb,
                                                                       nullptr, logits);
        vocab_softmax_kernel<<<BS, 256, 0, stream>>>(logits, sw, out, t);
        scatter_kernel<<<(BS * Ss + 255) / 256, 256, 0, stream>>>(prob, story, sw, out, t);
    }
}